// WeightController_2405181685868
// MI455X (gfx1250) — compile-verified
//
#include <hip/hip_runtime.h>

// ---------------------------------------------------------------------------
// PI-controller update (2048 lanes, negligible FLOPs) + 327.68 MB history
// shift. Pure HBM streaming: ~655 MB traffic -> ~28 us at 23.3 TB/s.
// CDNA5 path: async global<->LDS B128 copies (ASYNCcnt) with NT cache hints,
// double-buffered 4 KB tiles per wave, pointwise work merged into block 0.
// ---------------------------------------------------------------------------

#define N_BATCH    2048
#define N_HIST     8000
#define N_VARS     5
#define ROW        (N_BATCH * N_VARS)            // 10240 floats per history row
#define COPY_BYTES ((unsigned long long)(N_HIST - 1) * ROW * 4ull) // 327,639,040
#define TILE_BYTES 4096u                         // per-wave staging tile
#define SLOTS      8                             // 8 x 512B async chunks per tile
#define N_TILES    (COPY_BYTES / TILE_BYTES)     // exactly 79,990
#define BLOCKS     1024
#define TPB        256

// Issue 8 async B128 loads: global tile 'tidx' -> LDS buffer 'bufsel'
#define ASYNC_LOAD_TILE(bufsel, tidx)                                           \
  {                                                                             \
    unsigned long long _b = (tidx) * (unsigned long long)TILE_BYTES             \
                          + (unsigned long long)(lane * 16u);                   \
    _Pragma("unroll")                                                           \
    for (int _s = 0; _s < SLOTS; ++_s) {                                        \
      unsigned long long _g = srcb + _b + (unsigned)(_s * 512);                 \
      unsigned _l = lds_thread_base + (bufsel) * TILE_BYTES                     \
                  + (unsigned)(_s * 512);                                       \
      asm volatile("global_load_async_to_lds_b128 %0, %1, off th:TH_LOAD_NT"    \
                   :: "v"(_l), "v"(_g) : "memory");                             \
    }                                                                           \
  }

// Issue 8 async B128 stores: LDS buffer 'bufsel' -> global tile 'tidx'
#define ASYNC_STORE_TILE(bufsel, tidx)                                          \
  {                                                                             \
    unsigned long long _b = (tidx) * (unsigned long long)TILE_BYTES             \
                          + (unsigned long long)(lane * 16u);                   \
    _Pragma("unroll")                                                           \
    for (int _s = 0; _s < SLOTS; ++_s) {                                        \
      unsigned long long _g = dstb + _b + (unsigned)(_s * 512);                 \
      unsigned _l = lds_thread_base + (bufsel) * TILE_BYTES                     \
                  + (unsigned)(_s * 512);                                       \
      asm volatile("global_store_async_from_lds_b128 %0, %1, off th:TH_STORE_NT"\
                   :: "v"(_g), "v"(_l) : "memory");                             \
    }                                                                           \
  }

__global__ __launch_bounds__(TPB) void wc_fused_kernel(
        const float* __restrict__ cv,    // controlled_variable (2048)
        const float* __restrict__ ws,    // weight_state        (2048)
        const float* __restrict__ src,   // recording + ROW  (history row 1)
        float* __restrict__ out,         // full output base
        float* __restrict__ dst) {       // out + N_BATCH    (history row 0)
    __shared__ unsigned char lds_buf[8 * 2 * TILE_BYTES]; // 8 waves * 2 * 4KB = 64KB

    // ---------------- pointwise controller update (block 0 only) -----------
    if (blockIdx.x == 0) {
        for (int i = threadIdx.x; i < N_BATCH; i += TPB) {
            float c   = cv[i];
            float e   = -0.2f - c;                       // REFERENCE - cv
            float ti  = 4.0f * e;                        // K_I * e
            float trl = fminf(fmaxf(ti, -0.02f), 0.01f); // rate limit
            float wn  = fminf(fmaxf(ws[i] + trl, 0.0f), 1.5f);
            out[i] = wn;                                 // weight_state_next
            float* rec = out + N_BATCH
                       + (unsigned long long)(N_HIST - 1) * ROW
                       + (unsigned long long)i * N_VARS;
            rec[0] = c;
            rec[1] = wn;
            rec[2] = ti;
            rec[3] = trl;
            rec[4] = e;
        }
    }

    // ---------------- double-buffered async-LDS bulk shift -----------------
    const unsigned lane        = threadIdx.x & 31u;
    const unsigned wave_in_blk = threadIdx.x >> 5;
    const unsigned waves_total = (gridDim.x * blockDim.x) >> 5;
    const unsigned wave_id     = (blockIdx.x * blockDim.x + threadIdx.x) >> 5;

    // Flat LDS address truncated to 32 bits == DS-usable byte offset.
    const unsigned lds_thread_base = (unsigned)(unsigned long long)
        (const void*)&lds_buf[wave_in_blk * (2u * TILE_BYTES) + lane * 16u];

    const unsigned long long srcb = (unsigned long long)src;
    const unsigned long long dstb = (unsigned long long)dst;

    unsigned long long t = wave_id;
    if (t < N_TILES) {
        ASYNC_LOAD_TILE(0u, t);                    // prologue: fill buffer 0
        unsigned buf = 0u;
        for (; t < N_TILES; t += waves_total) {
            unsigned long long nxt = t + waves_total;
            // all prior ops done: buf's load complete, buf^1's store complete
            asm volatile("s_wait_asynccnt 0x0" ::: "memory");
            ASYNC_STORE_TILE(buf, t);              // drain current tile
            if (nxt < N_TILES)
                ASYNC_LOAD_TILE(buf ^ 1u, nxt);    // prefetch next tile
            buf ^= 1u;
        }
        asm volatile("s_wait_asynccnt 0x0" ::: "memory"); // drain final stores
    }
}

// ------------------------------- launcher -----------------------------------
extern "C" void kernel_launch(void* const* d_in, const int* in_sizes, int n_in,
                              void* d_out, int out_size, void* d_ws, size_t ws_size,
                              hipStream_t stream) {
    (void)in_sizes; (void)n_in; (void)out_size; (void)d_ws; (void)ws_size;
    const float* cv  = (const float*)d_in[0];   // controlled_variable (2048)
    const float* ws  = (const float*)d_in[1];   // weight_state        (2048)
    const float* rec = (const float*)d_in[2];   // recording (8000*2048*5)
    float* out = (float*)d_out;                 // [ws_next | recording_next]

    // single fused launch: rows 1..7999 -> output rows 0..7998 (79,990 tiles)
    // + controller update & last-row record in block 0
    wc_fused_kernel<<<BLOCKS, TPB, 0, stream>>>(cv, ws, rec + ROW, out,
                                                out + N_BATCH);
}